// Graph_Class_10041633538621
// MI455X (gfx1250) — compile-verified
//
#include <hip/hip_runtime.h>
#include <hip/hip_bf16.h>
#include <math.h>

#define NN 50000
#define EE 800000
#define FF 128
#define HH 128
#define NHEADS 4
#define GG 512
#define BN_EPS 1e-5f

typedef __attribute__((ext_vector_type(2))) float v2f;
typedef __attribute__((ext_vector_type(8))) float v8f;

// ---------- helpers ----------
__device__ __forceinline__ unsigned fkey(float f) {
    unsigned u = __float_as_uint(f);
    return (u & 0x80000000u) ? ~u : (u | 0x80000000u);
}
__device__ __forceinline__ float fdecode(unsigned u) {
    return __uint_as_float((u & 0x80000000u) ? (u ^ 0x80000000u) : ~u);
}
__device__ __forceinline__ float lrelu(float x) { return x > 0.f ? x : 0.2f * x; }
__device__ __forceinline__ float bn_eval(float v, float g, float b, float m, float var) {
    return (v - m) * rsqrtf(var + BN_EPS) * g + b;
}

// ---------- fp32 WMMA GEMM: C[M,Nc] = A[M,K] * B[K,Nc], all row-major ----------
// Register-blocked: one wave computes a 16x64 tile (4 independent 16x16 accumulators)
// via V_WMMA_F32_16X16X4_F32. A fragment loaded once per k-step, shared by 4 wmma.
// Requires M % 16 == 0, Nc % 64 == 0.
__global__ void wmma_gemm_f32(const float* __restrict__ A, const float* __restrict__ B,
                              float* __restrict__ C, int M, int K, int Nc) {
    const int lane = threadIdx.x & 31;
    const int wave = threadIdx.x >> 5;
    const int groupsN = Nc >> 6;                       // 64-col groups
    const int totalJobs = (M >> 4) * groupsN;
    const int job = blockIdx.x * (blockDim.x >> 5) + wave;
    if (job >= totalJobs) return;                      // wave-uniform, EXEC stays all-ones
    const int row0 = (job / groupsN) << 4;
    const int col0 = (job % groupsN) << 6;
    const int r  = lane & 15;
    const int kh = (lane >> 4) << 1;                   // 0 or 2 (K sub-offset per half-wave)
    v8f acc0 = {0.f,0.f,0.f,0.f,0.f,0.f,0.f,0.f};
    v8f acc1 = acc0, acc2 = acc0, acc3 = acc0;
    const float* __restrict__ arow = A + (size_t)(row0 + r) * K;
    for (int k = 0; k < K; k += 4) {
        v2f a;
        a[0] = arow[k + kh];
        a[1] = arow[k + kh + 1];
        const float* __restrict__ b0 = B + (size_t)(k + kh) * Nc + col0 + r;
        const float* __restrict__ b1 = B + (size_t)(k + kh + 1) * Nc + col0 + r;
        v2f bb0, bb1, bb2, bb3;
        bb0[0] = b0[0];  bb0[1] = b1[0];
        bb1[0] = b0[16]; bb1[1] = b1[16];
        bb2[0] = b0[32]; bb2[1] = b1[32];
        bb3[0] = b0[48]; bb3[1] = b1[48];
        acc0 = __builtin_amdgcn_wmma_f32_16x16x4_f32(false, a, false, bb0, (short)0, acc0, false, false);
        acc1 = __builtin_amdgcn_wmma_f32_16x16x4_f32(false, a, false, bb1, (short)0, acc1, false, false);
        acc2 = __builtin_amdgcn_wmma_f32_16x16x4_f32(false, a, false, bb2, (short)0, acc2, false, false);
        acc3 = __builtin_amdgcn_wmma_f32_16x16x4_f32(false, a, false, bb3, (short)0, acc3, false, false);
    }
    const int rowBase = row0 + ((lane >> 4) << 3);     // rows i (lanes 0-15) / i+8 (16-31)
#pragma unroll
    for (int i = 0; i < 8; ++i) {
        float* crow = C + (size_t)(rowBase + i) * Nc + col0 + r;
        crow[0]  = acc0[i];
        crow[16] = acc1[i];
        crow[32] = acc2[i];
        crow[48] = acc3[i];
    }
}

// ---------- init: pooling accumulators + degree ----------
__global__ void k_init(float* sadd, unsigned* smaxu, float* cnt, float* deg) {
    int t = blockIdx.x * blockDim.x + threadIdx.x;
    if (t < GG * 512) { sadd[t] = 0.f; smaxu[t] = 0x80000000u; }  // key(0.0f)
    if (t < GG) cnt[t] = 0.f;
    if (t < NN) deg[t] = 1.f;                 // self-loop contribution
}

__global__ void k_deg(const int* __restrict__ ei, float* deg) {
    int e = blockIdx.x * blockDim.x + threadIdx.x;
    if (e >= EE) return;
    atomicAdd(&deg[ei[EE + e]], 1.f);
}

__global__ void k_dis(float* deg) {          // in-place: dis = rsqrt(deg), deg >= 1
    int n = blockIdx.x * blockDim.x + threadIdx.x;
    if (n < NN) deg[n] = rsqrtf(deg[n]);
}

// hacc[n,:] = h0[n,:] * dis[n]^2   (self-loop term)
__global__ void k_hacc_init(const float* __restrict__ h0, const float* __restrict__ dis,
                            float* hacc) {
    int t = blockIdx.x * blockDim.x + threadIdx.x;
    if (t >= NN * HH) return;
    float d = dis[t >> 7];
    hacc[t] = h0[t] * d * d;
}

// GCN edge aggregation: one wave per edge, float4 per lane (32*4 = 128 feats)
__global__ void k_gcn_edge(const int* __restrict__ ei, const float* __restrict__ h0,
                           const float* __restrict__ dis, float* hacc) {
    int gw = (blockIdx.x * blockDim.x + threadIdx.x) >> 5;
    int lane = threadIdx.x & 31;
    if (gw >= EE) return;
    int s = ei[gw], d = ei[EE + gw];
    float w = dis[s] * dis[d];
    float4 v = ((const float4*)(h0 + (size_t)s * HH))[lane];
    float* dp = hacc + (size_t)d * HH + lane * 4;
    atomicAdd(dp + 0, v.x * w);
    atomicAdd(dp + 1, v.y * w);
    atomicAdd(dp + 2, v.z * w);
    atomicAdd(dp + 3, v.w * w);
}

// h = relu(bn2(hacc + b_gcn))
__global__ void k_bn_relu1(const float* __restrict__ hacc, const float* __restrict__ b_gcn,
                           const float* __restrict__ g, const float* __restrict__ b,
                           const float* __restrict__ m, const float* __restrict__ var,
                           float* h) {
    int t = blockIdx.x * blockDim.x + threadIdx.x;
    if (t >= NN * HH) return;
    int f = t & (HH - 1);
    float v = bn_eval(hacc[t] + b_gcn[f], g[f], b[f], m[f], var[f]);
    h[t] = v > 0.f ? v : 0.f;
}

// asrc[n,hd] = <hg[n,hd,:], a_src[hd,:]>, likewise adst  (float4 vectorized)
__global__ void k_attn_coef(const float* __restrict__ hg, const float* __restrict__ avs,
                            const float* __restrict__ avd, float* asrc, float* adst) {
    int t = blockIdx.x * blockDim.x + threadIdx.x;
    if (t >= NN * NHEADS) return;
    int hd = t & 3;
    const float4* row = (const float4*)(hg + ((size_t)(t >> 2) * 512) + hd * HH);
    const float4* as  = (const float4*)(avs + hd * HH);
    const float4* ad  = (const float4*)(avd + hd * HH);
    float sa = 0.f, sd = 0.f;
#pragma unroll 8
    for (int f = 0; f < HH / 4; ++f) {
        float4 v = row[f], s = as[f], d = ad[f];
        sa += v.x * s.x + v.y * s.y + v.z * s.z + v.w * s.w;
        sd += v.x * d.x + v.y * d.y + v.z * d.z + v.w * d.w;
    }
    asrc[t] = sa; adst[t] = sd;
}

__global__ void k_emax_init(const float* __restrict__ asrc, const float* __restrict__ adst,
                            unsigned* emaxu) {
    int t = blockIdx.x * blockDim.x + threadIdx.x;
    if (t >= NN * NHEADS) return;
    emaxu[t] = fkey(lrelu(asrc[t] + adst[t]));   // self-loop edge
}

__global__ void k_emax_edge(const int* __restrict__ ei, const float* __restrict__ asrc,
                            const float* __restrict__ adst, unsigned* emaxu) {
    int e = blockIdx.x * blockDim.x + threadIdx.x;
    if (e >= EE) return;
    int s = ei[e], d = ei[EE + e];
#pragma unroll
    for (int hd = 0; hd < NHEADS; ++hd) {
        float ev = lrelu(asrc[s * 4 + hd] + adst[d * 4 + hd]);
        atomicMax(&emaxu[d * 4 + hd], fkey(ev));
    }
}

__global__ void k_esum_init(const float* __restrict__ asrc, const float* __restrict__ adst,
                            const unsigned* __restrict__ emaxu, float* esum) {
    int t = blockIdx.x * blockDim.x + threadIdx.x;
    if (t >= NN * NHEADS) return;
    float es = lrelu(asrc[t] + adst[t]);
    esum[t] = expf(es - fdecode(emaxu[t]));
}

__global__ void k_esum_edge(const int* __restrict__ ei, const float* __restrict__ asrc,
                            const float* __restrict__ adst, const unsigned* __restrict__ emaxu,
                            float* esum) {
    int e = blockIdx.x * blockDim.x + threadIdx.x;
    if (e >= EE) return;
    int s = ei[e], d = ei[EE + e];
#pragma unroll
    for (int hd = 0; hd < NHEADS; ++hd) {
        float ev = lrelu(asrc[s * 4 + hd] + adst[d * 4 + hd]);
        atomicAdd(&esum[d * 4 + hd], expf(ev - fdecode(emaxu[d * 4 + hd])));
    }
}

// outacc[n,f] = alpha_self(n,head(f)) * hg[n,f]
__global__ void k_out_init(const float* __restrict__ hg, const float* __restrict__ asrc,
                           const float* __restrict__ adst, const unsigned* __restrict__ emaxu,
                           const float* __restrict__ esum, float* outacc) {
    size_t t = (size_t)blockIdx.x * blockDim.x + threadIdx.x;
    if (t >= (size_t)NN * 512) return;
    int n = (int)(t >> 9), f = (int)(t & 511), hd = f >> 7;
    int idx = n * 4 + hd;
    float es = lrelu(asrc[idx] + adst[idx]);
    float alpha = expf(es - fdecode(emaxu[idx])) / (esum[idx] + 1e-16f);
    outacc[t] = alpha * hg[t];
}

// GAT edge scatter: wave per edge (grid-strided), prefetch next source row
__global__ void k_gat_edge(const int* __restrict__ ei, const float* __restrict__ hg,
                           const float* __restrict__ asrc, const float* __restrict__ adst,
                           const unsigned* __restrict__ emaxu, const float* __restrict__ esum,
                           float* outacc, int nwaves) {
    int gw = (blockIdx.x * blockDim.x + threadIdx.x) >> 5;
    int lane = threadIdx.x & 31;
    for (int e = gw; e < EE; e += nwaves) {
        int nxt = e + nwaves;
        if (nxt < EE) {  // prefetch next iteration's 2KB source row (64B/lane)
            const float* p = hg + (size_t)ei[nxt] * 512 + lane * 16;
            __builtin_prefetch(p, 0, 1);
        }
        int s = ei[e], d = ei[EE + e];
#pragma unroll
        for (int hd = 0; hd < NHEADS; ++hd) {
            int idx = d * 4 + hd;
            float ev = lrelu(asrc[s * 4 + hd] + adst[idx]);
            float alpha = expf(ev - fdecode(emaxu[idx])) / (esum[idx] + 1e-16f);
            float4 v = ((const float4*)(hg + (size_t)s * 512 + hd * HH))[lane];
            float* dp = outacc + (size_t)d * 512 + hd * HH + lane * 4;
            atomicAdd(dp + 0, v.x * alpha);
            atomicAdd(dp + 1, v.y * alpha);
            atomicAdd(dp + 2, v.z * alpha);
            atomicAdd(dp + 3, v.w * alpha);
        }
    }
}

// BN1 + ReLU + pooled add/max
__global__ void k_pool(const float* __restrict__ outacc, const float* __restrict__ b_gat,
                       const float* __restrict__ g, const float* __restrict__ b,
                       const float* __restrict__ m, const float* __restrict__ var,
                       const int* __restrict__ batch, float* sadd, unsigned* smaxu) {
    size_t t = (size_t)blockIdx.x * blockDim.x + threadIdx.x;
    if (t >= (size_t)NN * 512) return;
    int n = (int)(t >> 9), f = (int)(t & 511);
    float v = bn_eval(outacc[t] + b_gat[f], g[f], b[f], m[f], var[f]);
    v = v > 0.f ? v : 0.f;
    int gidx = batch[n];
    atomicAdd(&sadd[(size_t)gidx * 512 + f], v);
    atomicMax(&smaxu[(size_t)gidx * 512 + f], fkey(v));
}

__global__ void k_cnt(const int* __restrict__ batch, float* cnt) {
    int n = blockIdx.x * blockDim.x + threadIdx.x;
    if (n < NN) atomicAdd(&cnt[batch[n]], 1.f);
}

// out[g,j] = sigmoid( [sadd|smax|smean] . w_lin[:,j] + b_lin[j] )
__global__ void k_final(const float* __restrict__ sadd, const unsigned* __restrict__ smaxu,
                        const float* __restrict__ cnt, const float* __restrict__ w_lin,
                        const float* __restrict__ b_lin, float* out) {
    int t = blockIdx.x * blockDim.x + threadIdx.x;
    if (t >= GG * 16) return;
    int gidx = t >> 4, j = t & 15;
    float c = fmaxf(cnt[gidx], 1.f);
    float acc = b_lin[j];
    for (int k = 0; k < 512; ++k) {
        float sa = sadd[(size_t)gidx * 512 + k];
        float sm = fdecode(smaxu[(size_t)gidx * 512 + k]);
        float me = sa / c;
        acc += sa * w_lin[k * 16 + j];
        acc += sm * w_lin[(512 + k) * 16 + j];
        acc += me * w_lin[(1024 + k) * 16 + j];
    }
    out[t] = 1.f / (1.f + expf(-acc));
}

// ---------- host launcher ----------
static inline size_t alignUp(size_t x) { return (x + 255) & ~(size_t)255; }

extern "C" void kernel_launch(void* const* d_in, const int* in_sizes, int n_in,
                              void* d_out, int out_size, void* d_ws, size_t ws_size,
                              hipStream_t stream) {
    const float* x      = (const float*)d_in[0];
    const int*   ei     = (const int*)d_in[1];
    const int*   batch  = (const int*)d_in[2];
    const float* w_gcn  = (const float*)d_in[3];
    const float* b_gcn  = (const float*)d_in[4];
    const float* g2g    = (const float*)d_in[5];
    const float* g2b    = (const float*)d_in[6];
    const float* g2m    = (const float*)d_in[7];
    const float* g2v    = (const float*)d_in[8];
    const float* w_gat  = (const float*)d_in[9];
    const float* a_src  = (const float*)d_in[10];
    const float* a_dst  = (const float*)d_in[11];
    const float* b_gat  = (const float*)d_in[12];
    const float* g1g    = (const float*)d_in[13];
    const float* g1b    = (const float*)d_in[14];
    const float* g1m    = (const float*)d_in[15];
    const float* g1v    = (const float*)d_in[16];
    const float* w_lin  = (const float*)d_in[17];
    const float* b_lin  = (const float*)d_in[18];
    float* out = (float*)d_out;

    char* ws = (char*)d_ws;
    size_t off = 0;
    float*    h0     = (float*)(ws + off); off = alignUp(off + (size_t)NN * HH * 4);
    float*    hacc   = (float*)(ws + off); off = alignUp(off + (size_t)NN * HH * 4);
    float*    dis    = (float*)(ws + off); off = alignUp(off + (size_t)NN * 4);
    float*    hg     = (float*)(ws + off); off = alignUp(off + (size_t)NN * 512 * 4);
    float*    asrc   = (float*)(ws + off); off = alignUp(off + (size_t)NN * 4 * 4);
    float*    adst   = (float*)(ws + off); off = alignUp(off + (size_t)NN * 4 * 4);
    unsigned* emaxu  = (unsigned*)(ws + off); off = alignUp(off + (size_t)NN * 4 * 4);
    float*    esum   = (float*)(ws + off); off = alignUp(off + (size_t)NN * 4 * 4);
    float*    outacc = (float*)(ws + off); off = alignUp(off + (size_t)NN * 512 * 4);
    float*    sadd   = (float*)(ws + off); off = alignUp(off + (size_t)GG * 512 * 4);
    unsigned* smaxu  = (unsigned*)(ws + off); off = alignUp(off + (size_t)GG * 512 * 4);
    float*    cnt    = (float*)(ws + off); off = alignUp(off + (size_t)GG * 4);

    const int T = 256;
    auto cdiv = [](long long a, long long b) { return (int)((a + b - 1) / b); };

    // init pooling accumulators + degree (self-loops)
    k_init<<<cdiv((long long)GG * 512, T), T, 0, stream>>>(sadd, smaxu, cnt, dis /*deg*/);
    // GCN: h0 = x @ w_gcn   (WMMA fp32, 16x64 per wave -> 6250 wave jobs)
    wmma_gemm_f32<<<cdiv((NN / 16) * (HH / 64), 8), T, 0, stream>>>(x, w_gcn, h0, NN, FF, HH);
    // degree + sym norm
    k_deg<<<cdiv(EE, T), T, 0, stream>>>(ei, dis);
    k_dis<<<cdiv(NN, T), T, 0, stream>>>(dis);
    // self-loop term + edge aggregation
    k_hacc_init<<<cdiv((long long)NN * HH, T), T, 0, stream>>>(h0, dis, hacc);
    k_gcn_edge<<<cdiv((long long)EE * 32, T), T, 0, stream>>>(ei, h0, dis, hacc);
    // BN2 + ReLU -> reuse h0 buffer as h
    k_bn_relu1<<<cdiv((long long)NN * HH, T), T, 0, stream>>>(hacc, b_gcn, g2g, g2b, g2m, g2v, h0);
    // GAT: hg = h @ w_gat   (WMMA fp32, 16x64 per wave -> 25000 wave jobs)
    wmma_gemm_f32<<<cdiv((NN / 16) * (512 / 64), 8), T, 0, stream>>>(h0, w_gat, hg, NN, HH, NHEADS * HH);
    // attention coefficients + segment softmax
    k_attn_coef<<<cdiv((long long)NN * NHEADS, T), T, 0, stream>>>(hg, a_src, a_dst, asrc, adst);
    k_emax_init<<<cdiv((long long)NN * NHEADS, T), T, 0, stream>>>(asrc, adst, emaxu);
    k_emax_edge<<<cdiv(EE, T), T, 0, stream>>>(ei, asrc, adst, emaxu);
    k_esum_init<<<cdiv((long long)NN * NHEADS, T), T, 0, stream>>>(asrc, adst, emaxu, esum);
    k_esum_edge<<<cdiv(EE, T), T, 0, stream>>>(ei, asrc, adst, emaxu, esum);
    // weighted aggregation (self-loop init + edge scatter with prefetch)
    k_out_init<<<cdiv((long long)NN * 512, T), T, 0, stream>>>(hg, asrc, adst, emaxu, esum, outacc);
    const int gatBlocks = 4096;
    k_gat_edge<<<gatBlocks, T, 0, stream>>>(ei, hg, asrc, adst, emaxu, esum, outacc, gatBlocks * (T / 32));
    // BN1 + ReLU + pooling
    k_pool<<<cdiv((long long)NN * 512, T), T, 0, stream>>>(outacc, b_gat, g1g, g1b, g1m, g1v, batch, sadd, smaxu);
    k_cnt<<<cdiv(NN, T), T, 0, stream>>>(batch, cnt);
    // final linear + sigmoid
    k_final<<<cdiv(GG * 16, T), T, 0, stream>>>(sadd, smaxu, cnt, w_lin, b_lin, out);
    (void)in_sizes; (void)n_in; (void)out_size; (void)ws_size;
}